// MultiBoxLoss_14396730376697
// MI455X (gfx1250) — compile-verified
//
#include <hip/hip_runtime.h>
#include <stdint.h>

// ---------------- problem constants (from reference) ----------------
#define BATCH        64
#define PRIORS       16800
#define NEG_POS_RATIO 7
#define K1_THREADS   256
#define K1_BLOCKS    ((BATCH * PRIORS) / K1_THREADS)     // 4200, exact
#define K2_THREADS   1024
#define MAXLOC       ((PRIORS + K2_THREADS - 1) / K2_THREADS) // 17

typedef float v2f  __attribute__((ext_vector_type(2)));
typedef float v8f  __attribute__((ext_vector_type(8)));
typedef unsigned int u32x4 __attribute__((ext_vector_type(4)));
typedef int   i32x4 __attribute__((ext_vector_type(4)));
typedef int   i32x8 __attribute__((ext_vector_type(8)));

#if defined(__gfx1250__) && __has_builtin(__builtin_amdgcn_tensor_load_to_lds) && __has_builtin(__builtin_amdgcn_s_wait_tensorcnt)
#define HAVE_TDM 1
#endif

__device__ __forceinline__ float smooth_l1(float d) {
  d = fabsf(d);
  return (d < 1.0f) ? 0.5f * d * d : d - 0.5f;
}

// ---------------- TDM helper: 1D DMA of nelem f32/i32 (global -> LDS) -------
#ifdef HAVE_TDM
__device__ __forceinline__ void tdm_load_1d(const void* gsrc, unsigned lds_off, unsigned nelem) {
  uint64_t ga = (uint64_t)(uintptr_t)gsrc;
  // D# group 0: count=1, lds_addr, global_addr[56:0], type=2 (bits 127:126)
  u32x4 g0;
  g0[0] = 1u;
  g0[1] = lds_off;
  g0[2] = (unsigned)(ga & 0xFFFFFFFFu);
  g0[3] = (unsigned)((ga >> 32) & 0x01FFFFFFu) | (2u << 30);
  // D# group 1: data_size=4B, tensor_dim0=nelem, tensor_dim1=1, tile_dim0=nelem,
  //             tile_dim1/2=0 (1D), tensor_dim0_stride=nelem
  i32x8 g1;
  g1[0] = (int)(2u << 16);                               // data_size = 4 bytes
  g1[1] = (int)((nelem & 0xFFFFu) << 16);                // tensor_dim0[15:0]
  g1[2] = (int)(((nelem >> 16) & 0xFFFFu) | (1u << 16)); // tensor_dim0[31:16], tensor_dim1=1
  g1[3] = (int)((nelem & 0xFFFFu) << 16);                // tile_dim0
  g1[4] = 0;                                             // tile_dim1=0, tile_dim2=0
  g1[5] = (int)nelem;                                    // tensor_dim0_stride[31:0]
  g1[6] = 0;
  g1[7] = 0;
  i32x4 z4 = {0, 0, 0, 0};
#if __clang_major__ >= 23
  i32x8 z8 = {0, 0, 0, 0, 0, 0, 0, 0};
  __builtin_amdgcn_tensor_load_to_lds(g0, g1, z4, z4, z8, 0);
#else
  __builtin_amdgcn_tensor_load_to_lds(g0, g1, z4, z4, 0);
#endif
}
#endif

// ---------------- Kernel 1: smooth-L1 sums (bulk 120 MB stream) -------------
// One block = 256 consecutive priors (flat over B*P; masks are per-prior).
// LDS layout (single symbol -> LDS offset 0):
#define LOC_OFF   0u
#define LOCT_OFF  4096u
#define LM_OFF    8192u
#define LMT_OFF   18432u
#define CT_OFF    28672u
#define RED_OFF   29696u
#define SMEM_TOT  30720u

__global__ __launch_bounds__(K1_THREADS) void k1_smoothl1(
    const float* __restrict__ loc_data, const float* __restrict__ loc_t,
    const float* __restrict__ landm_data, const float* __restrict__ landm_t,
    const int* __restrict__ conf_t, float* __restrict__ partials) {
  __shared__ __align__(16) unsigned char smem[SMEM_TOT];
  float* s_loc  = (float*)(smem + LOC_OFF);
  float* s_loct = (float*)(smem + LOCT_OFF);
  float* s_lm   = (float*)(smem + LM_OFF);
  float* s_lmt  = (float*)(smem + LMT_OFF);
  int*   s_ct   = (int*)(smem + CT_OFF);
  float* s_red  = (float*)(smem + RED_OFF);

  const int t = threadIdx.x;
  const long long g0 = (long long)blockIdx.x * K1_THREADS;   // first flat prior

#ifdef HAVE_TDM
  if (t == 0) {
    tdm_load_1d(loc_data   + g0 * 4,  LOC_OFF,  1024u);
    tdm_load_1d(loc_t      + g0 * 4,  LOCT_OFF, 1024u);
    tdm_load_1d(landm_data + g0 * 10, LM_OFF,   2560u);
    tdm_load_1d(landm_t    + g0 * 10, LMT_OFF,  2560u);
    tdm_load_1d((const void*)(conf_t + g0), CT_OFF, 256u);
    __builtin_amdgcn_s_wait_tensorcnt(0);
  }
  __syncthreads();
#else
  {
    const float4* a = (const float4*)(loc_data + g0 * 4);
    const float4* b = (const float4*)(loc_t + g0 * 4);
    ((float4*)s_loc)[t]  = a[t];
    ((float4*)s_loct)[t] = b[t];
    const float4* c = (const float4*)(landm_data + g0 * 10);
    const float4* d = (const float4*)(landm_t + g0 * 10);
    float4* sc = (float4*)s_lm;
    float4* sd = (float4*)s_lmt;
    for (int i = t; i < 640; i += K1_THREADS) { sc[i] = c[i]; sd[i] = d[i]; }
    if (t < 64) ((float4*)s_ct)[t] = ((const float4*)(conf_t + g0))[t];
  }
  __syncthreads();
#endif

  const int ct = s_ct[t];
  const float ml  = (ct != 0) ? 1.0f : 0.0f;   // pos_loc mask
  const float mlm = (ct > 0)  ? 1.0f : 0.0f;   // pos_landm mask
  float accL = 0.0f, accM = 0.0f;
#pragma unroll
  for (int j = 0; j < 4; ++j)  accL += smooth_l1(s_loc[t * 4 + j]  - s_loct[t * 4 + j]);
#pragma unroll
  for (int j = 0; j < 10; ++j) accM += smooth_l1(s_lm[t * 10 + j]  - s_lmt[t * 10 + j]);
  accL *= ml;
  accM *= mlm;

  // deterministic tree reductions
  s_red[t] = accL;
  __syncthreads();
  for (int s = K1_THREADS / 2; s > 0; s >>= 1) {
    if (t < s) s_red[t] += s_red[t + s];
    __syncthreads();
  }
  if (t == 0) partials[blockIdx.x * 2 + 0] = s_red[0];
  __syncthreads();
  s_red[t] = accM;
  __syncthreads();
  for (int s = K1_THREADS / 2; s > 0; s >>= 1) {
    if (t < s) s_red[t] += s_red[t + s];
    __syncthreads();
  }
  if (t == 0) partials[blockIdx.x * 2 + 1] = s_red[0];
}

// ---------------- Kernel 2: CE + exact top-k (radix select) per row ---------
__global__ __launch_bounds__(K2_THREADS) void k2_conf(
    const float* __restrict__ conf, const int* __restrict__ conf_t,
    float* __restrict__ row_cla, float* __restrict__ row_np,
    float* __restrict__ row_nplm) {
  __shared__ unsigned s_hist[256];
  __shared__ float    s_f[K2_THREADS];
  __shared__ unsigned s_u[K2_THREADS];
  __shared__ unsigned s_prefix;
  __shared__ int      s_kk;
  __shared__ int      s_k;
  __shared__ float    s_posce;
  __shared__ unsigned s_counts;

  const int b = blockIdx.x;
  const int t = threadIdx.x;
  const float* cf  = conf   + (size_t)b * PRIORS * 2;
  const int*   ctp = conf_t + (size_t)b * PRIORS;

  // per-thread mining values in VGPRs (fully unrolled; pad with +0.0 which is
  // mathematically inert for sum-of-top-k since all values are >= +0.0)
  float mv[MAXLOC];
  float posce = 0.0f;
  unsigned cnt = 0;  // nposloc | (nposlandm << 16)
#pragma unroll
  for (int i = 0; i < MAXLOC; ++i) {
    const int p = t + i * K2_THREADS;
    float m0 = 0.0f;
    if (p < PRIORS) {
      const float c0 = cf[2 * p], c1 = cf[2 * p + 1];
      const int ctv = ctp[p];
      const float mx  = fmaxf(c0, c1);
      const float lse = mx + log1pf(expf(-fabsf(c0 - c1)));
      const int isl = (ctv != 0);
      const float ce = lse - (isl ? c1 : c0);   // conf_t_loc = isl ? 1 : 0 (C=2)
      m0 = isl ? 0.0f : ce;                     // mining_loss
      posce += isl ? ce : 0.0f;
      cnt += (unsigned)isl + ((unsigned)(ctv > 0) << 16);
    }
    mv[i] = m0;
  }

  s_f[t] = posce; s_u[t] = cnt;
  __syncthreads();
  for (int s = K2_THREADS / 2; s > 0; s >>= 1) {
    if (t < s) { s_f[t] += s_f[t + s]; s_u[t] += s_u[t + s]; }
    __syncthreads();
  }
  if (t == 0) {
    s_posce  = s_f[0];
    s_counts = s_u[0];
    int np = (int)(s_u[0] & 0xFFFFu);
    int k  = NEG_POS_RATIO * np;
    if (k > PRIORS - 1) k = PRIORS - 1;
    s_k = k; s_kk = k; s_prefix = 0u;
  }
  __syncthreads();

  const int k = s_k;
  float cla = s_posce;
  if (k > 0) {
    // 4-round 8-bit radix select for the k-th largest mining value.
    // Non-negative floats => uint bit order == value order.
    for (int r = 3; r >= 0; --r) {
      if (t < 256) s_hist[t] = 0u;
      __syncthreads();
      const unsigned shift  = 8u * (unsigned)r;
      const unsigned himask = (r == 3) ? 0u : (0xFFFFFFFFu << (8u * (unsigned)(r + 1)));
      const unsigned pref   = s_prefix;
#pragma unroll
      for (int i = 0; i < MAXLOC; ++i) {
        const unsigned u = __float_as_uint(mv[i]);
        if ((u & himask) == pref) atomicAdd(&s_hist[(u >> shift) & 255u], 1u);
      }
      __syncthreads();
      if (t == 0) {
        unsigned cum = 0;
        int kk = s_kk;
        for (int d = 255; d >= 0; --d) {
          const unsigned h = s_hist[d];
          if (cum + h >= (unsigned)kk) {
            s_prefix = pref | ((unsigned)d << shift);
            s_kk = kk - (int)cum;
            break;
          }
          cum += h;
        }
      }
      __syncthreads();
    }
    const unsigned x = s_prefix;       // bits of the k-th largest value
    float sg = 0.0f; unsigned cg = 0;
#pragma unroll
    for (int i = 0; i < MAXLOC; ++i) {
      const unsigned u = __float_as_uint(mv[i]);
      if (u > x) { sg += mv[i]; cg++; }
    }
    s_f[t] = sg; s_u[t] = cg;
    __syncthreads();
    for (int s = K2_THREADS / 2; s > 0; s >>= 1) {
      if (t < s) { s_f[t] += s_f[t + s]; s_u[t] += s_u[t + s]; }
      __syncthreads();
    }
    if (t == 0) cla += s_f[0] + (float)(k - (int)s_u[0]) * __uint_as_float(x);
  }
  if (t == 0) {
    row_cla[b]  = cla;
    row_np[b]   = (float)(s_counts & 0xFFFFu);
    row_nplm[b] = (float)(s_counts >> 16);
  }
}

// ---------------- Kernel 3: final reductions + normalize --------------------
// WMMA 64-value wave reduction: A = ones(16x4), B = data(4x16) => D = colsums.
__device__ __forceinline__ float wave_sum64(const float* __restrict__ q) {
#if defined(__gfx1250__) && __has_builtin(__builtin_amdgcn_wmma_f32_16x16x4_f32)
  const int L  = (int)(threadIdx.x & 31u);
  const int i0 = (L < 16) ? L : (L + 16);
  v2f av; av[0] = 1.0f; av[1] = 1.0f;
  v2f bv; bv[0] = q[i0]; bv[1] = q[i0 + 16];
  v8f cz = {};
  v8f d = __builtin_amdgcn_wmma_f32_16x16x4_f32(false, av, false, bv, (short)0, cz,
                                                false, false);
  float s = d[0];                 // lane L holds colsum(L & 15)
  s += __shfl_xor(s, 8, 32);
  s += __shfl_xor(s, 4, 32);
  s += __shfl_xor(s, 2, 32);
  s += __shfl_xor(s, 1, 32);
  return s;
#else
  float s = 0.0f;
  for (int i = 0; i < 64; ++i) s += q[i];
  return s;
#endif
}

__global__ __launch_bounds__(256) void k3_final(
    const float* __restrict__ partials, const float* __restrict__ row_cla,
    const float* __restrict__ row_np, const float* __restrict__ row_nplm,
    float* __restrict__ out) {
  __shared__ float sA[256], sB[256];
  __shared__ float sh_loc, sh_lm;
  const int t = threadIdx.x;
  float a = 0.0f, c = 0.0f;
  for (int i = t; i < K1_BLOCKS; i += 256) {
    a += partials[2 * i + 0];
    c += partials[2 * i + 1];
  }
  sA[t] = a; sB[t] = c;
  __syncthreads();
  for (int s = 128; s > 0; s >>= 1) {
    if (t < s) { sA[t] += sA[t + s]; sB[t] += sB[t + s]; }
    __syncthreads();
  }
  if (t == 0) { sh_loc = sA[0]; sh_lm = sB[0]; }
  __syncthreads();
  if (t < 32) {   // wave 0, EXEC all ones -> WMMA legal
    const float cla = wave_sum64(row_cla);
    const float np  = wave_sum64(row_np);
    const float nlm = wave_sum64(row_nplm);
    if (t == 0) {
      const float N  = fmaxf(np, 1.0f);
      const float N1 = fmaxf(nlm, 1.0f);
      out[0] = sh_loc / N;
      out[1] = cla / N;
      out[2] = sh_lm / N1;
    }
  }
}

// ---------------- host entry ------------------------------------------------
extern "C" void kernel_launch(void* const* d_in, const int* in_sizes, int n_in,
                              void* d_out, int out_size, void* d_ws, size_t ws_size,
                              hipStream_t stream) {
  (void)in_sizes; (void)n_in; (void)out_size; (void)ws_size;
  const float* loc_data   = (const float*)d_in[0];
  const float* conf_data  = (const float*)d_in[1];
  const float* landm_data = (const float*)d_in[2];
  const float* loc_t      = (const float*)d_in[3];
  const float* landm_t    = (const float*)d_in[4];
  const int*   conf_t     = (const int*)d_in[5];
  float* out = (float*)d_out;

  float* ws       = (float*)d_ws;
  float* partials = ws;                          // 2 * 4200 floats
  float* row_cla  = ws + 2 * K1_BLOCKS;          // 64
  float* row_np   = row_cla + BATCH;             // 64
  float* row_nplm = row_np + BATCH;              // 64

  k1_smoothl1<<<K1_BLOCKS, K1_THREADS, 0, stream>>>(loc_data, loc_t, landm_data,
                                                    landm_t, conf_t, partials);
  k2_conf<<<BATCH, K2_THREADS, 0, stream>>>(conf_data, conf_t, row_cla, row_np,
                                            row_nplm);
  k3_final<<<1, 256, 0, stream>>>(partials, row_cla, row_np, row_nplm, out);
}